// EMA_17068200034909
// MI455X (gfx1250) — compile-verified
//
#include <hip/hip_runtime.h>
#include <cstdint>
#include <cstddef>

namespace {

constexpr int kB   = 8;
constexpr int kT   = 8192;
constexpr int kD   = 512;
constexpr int kTPB = 128;   // threads per block; each thread owns 4 consecutive channels
constexpr int kTS  = 8;     // timesteps staged per LDS tile (8 * 2KB = 16KB per buffer)

constexpr int ilog2_c(int v) { int r = 0; while (v > 1) { v >>= 1; ++r; } return r; }

typedef float v4f __attribute__((ext_vector_type(4)));

__device__ __forceinline__ float sigmoidf_(float t) {
  return 1.0f / (1.0f + __expf(-t));
}

// CDNA5 async DMA: copies 16 bytes from global address gaddr to LDS offset lds_off, tracked by ASYNCcnt.
__device__ __forceinline__ void async_ld4(unsigned lds_off, const void* gaddr) {
  unsigned long long ga = (unsigned long long)gaddr;
  asm volatile("global_load_async_to_lds_b128 %0, %1, off"
               :: "v"(lds_off), "v"(ga)
               : "memory");
}

// Wait until at most 8 async ops outstanding (the next tile's prefetch).
__device__ __forceinline__ void wait_async_le8() {
  asm volatile("s_wait_asynccnt 0x8" ::: "memory");
}
__device__ __forceinline__ void wait_async_0() {
  asm volatile("s_wait_asynccnt 0x0" ::: "memory");
}

// Non-temporal 16B store (lowers to global_store_b128 ... th:TH_STORE_NT):
// keeps the 134MB output stream from evicting x out of the 192MB L2 between passes.
__device__ __forceinline__ void store_nt4(float* p, float4 v) {
  v4f w; w.x = v.x; w.y = v.y; w.z = v.z; w.w = v.w;
  __builtin_nontemporal_store(w, (v4f*)p);
}

// Pass 1 (WRITE_OUT=false): scan chunk from h=0, write chunk-end carry to carry_or_init.
// Pass 3 (WRITE_OUT=true):  scan chunk from exact entry state (read from carry_or_init),
//                           stream h_t to out with NT stores.
template <int CHUNKS, bool WRITE_OUT>
__global__ __launch_bounds__(kTPB) void ema_scan_kernel(
    const float* __restrict__ x,
    const float* __restrict__ alpha_logit,
    float* __restrict__ carry_or_init,
    float* __restrict__ out) {
  constexpr int L  = kT / CHUNKS;   // timesteps per chunk
  constexpr int NT = L / kTS;       // LDS tiles per chunk

  __shared__ float4 tile[2][kTS * kTPB];   // 32 KB double buffer

  const int j  = threadIdx.x;
  const int d0 = j * 4;
  const int bc = blockIdx.x;
  const int b  = bc / CHUNKS;
  const int c  = bc % CHUNKS;

  const float4 lg = *(const float4*)(alpha_logit + d0);
  float4 a, om;
  a.x = sigmoidf_(lg.x); a.y = sigmoidf_(lg.y);
  a.z = sigmoidf_(lg.z); a.w = sigmoidf_(lg.w);
  om.x = 1.0f - a.x; om.y = 1.0f - a.y;
  om.z = 1.0f - a.z; om.w = 1.0f - a.w;

  const size_t base = ((size_t)b * kT + (size_t)c * L) * kD + (size_t)d0;
  const float* gx = x + base;

  // Low 32 bits of a flat pointer to LDS == LDS byte offset (shared aperture is 4GB-aligned).
  const unsigned lds_base = (unsigned)(uintptr_t)(&tile[0][0]);
  const unsigned slot     = (unsigned)j * 16u;

  float4 h;
  if (WRITE_OUT) {
    h = *(const float4*)(carry_or_init + ((size_t)b * CHUNKS + c) * kD + d0);
  } else {
    h = make_float4(0.f, 0.f, 0.f, 0.f);
  }

  // Prefetch tile 0 into buffer 0.
#pragma unroll
  for (int i = 0; i < kTS; ++i) {
    async_ld4(lds_base + (unsigned)(i * kTPB) * 16u + slot, gx + (size_t)i * kD);
  }

  for (int nt = 0; nt < NT; ++nt) {
    const int buf = nt & 1;
    if (nt + 1 < NT) {
      const unsigned lnext = lds_base + (unsigned)((buf ^ 1) * kTS * kTPB) * 16u;
      const float* gnext = gx + (size_t)(nt + 1) * kTS * kD;
#pragma unroll
      for (int i = 0; i < kTS; ++i) {
        async_ld4(lnext + (unsigned)(i * kTPB) * 16u + slot, gnext + (size_t)i * kD);
      }
      wait_async_le8();   // current tile's 8 loads complete (in-order), next 8 in flight
    } else {
      wait_async_0();
    }
#pragma unroll
    for (int i = 0; i < kTS; ++i) {
      const float4 v = tile[buf][i * kTPB + j];
      h.x = fmaf(om.x, v.x, a.x * h.x);
      h.y = fmaf(om.y, v.y, a.y * h.y);
      h.z = fmaf(om.z, v.z, a.z * h.z);
      h.w = fmaf(om.w, v.w, a.w * h.w);
      if (WRITE_OUT) {
        store_nt4(out + base + (size_t)(nt * kTS + i) * kD, h);
      }
    }
  }

  if (!WRITE_OUT) {
    // Regular-temporal store: pass 2 re-reads these within microseconds.
    *(float4*)(carry_or_init + ((size_t)b * CHUNKS + c) * kD + d0) = h;
  }
}

// Pass 2: per channel, exact in-place scan over chunk carries.
// Entry state of chunk c: H_c = a^L * H_{c-1} + carry_{c-1}; carry slot c is
// overwritten with H_c (entry state) for pass 3.
template <int CHUNKS>
__global__ void ema_chunkscan_kernel(const float* __restrict__ alpha_logit,
                                     float* __restrict__ carry) {
  constexpr int L     = kT / CHUNKS;
  constexpr int LOG2L = ilog2_c(L);
  const int ch = blockIdx.x * blockDim.x + threadIdx.x;
  if (ch >= kB * kD) return;
  const int b = ch / kD;
  const int d = ch - b * kD;

  const float a = sigmoidf_(alpha_logit[d]);
  float aL = a;
#pragma unroll
  for (int k = 0; k < LOG2L; ++k) aL *= aL;   // a^(2^LOG2L) == a^L

  float H = 0.f;
  float* s = carry + (size_t)b * CHUNKS * kD + d;
  for (int c = 0; c < CHUNKS; ++c) {
    const float sc = s[(size_t)c * kD];
    s[(size_t)c * kD] = H;          // entry state for chunk c
    H = fmaf(aL, H, sc);
  }
}

template <int CHUNKS>
void run_ema(const float* x, const float* lg, float* out, float* ws,
             hipStream_t stream) {
  dim3 grid(kB * CHUNKS), blk(kTPB);
  ema_scan_kernel<CHUNKS, false><<<grid, blk, 0, stream>>>(x, lg, ws, nullptr);
  const int nch = kB * kD;
  ema_chunkscan_kernel<CHUNKS><<<(nch + 255) / 256, 256, 0, stream>>>(lg, ws);
  ema_scan_kernel<CHUNKS, true><<<grid, blk, 0, stream>>>(x, lg, ws, out);
}

}  // namespace

extern "C" void kernel_launch(void* const* d_in, const int* in_sizes, int n_in,
                              void* d_out, int out_size, void* d_ws, size_t ws_size,
                              hipStream_t stream) {
  (void)in_sizes; (void)n_in; (void)out_size;
  const float* x  = (const float*)d_in[0];
  const float* lg = (const float*)d_in[1];
  float* out = (float*)d_out;
  float* ws  = (float*)d_ws;

  const auto need = [](int ch) {
    return (size_t)kB * (size_t)ch * (size_t)kD * sizeof(float);
  };

  if (ws_size >= need(128)) {        // 2 MB scratch: 1024 blocks, best latency hiding
    run_ema<128>(x, lg, out, ws, stream);
  } else if (ws_size >= need(64)) {  // 1 MB
    run_ema<64>(x, lg, out, ws, stream);
  } else if (ws_size >= need(16)) {  // 256 KB
    run_ema<16>(x, lg, out, ws, stream);
  } else {                           // 16 KB minimum
    run_ema<1>(x, lg, out, ws, stream);
  }
}